// AlignBlock_22213570855321
// MI455X (gfx1250) — compile-verified
//
#include <hip/hip_runtime.h>
#include <hip/hip_bf16.h>
#include <math.h>

// Problem constants (fixed by reference setup_inputs)
#define B_  2
#define C_  2
#define T_  500
#define F_  129
#define H_  16
#define D_  100
#define FP_ 132   // F padded to multiple of 4 for K=4 WMMA steps
#define TP_ 512   // T padded to multiple of 16 (rows 500..511 zeroed)
#define NF_ (FP_ / 4)   // 33 K=4 chunks

typedef float v2f __attribute__((ext_vector_type(2)));
typedef float v8f __attribute__((ext_vector_type(8)));

// ---------------------------------------------------------------------------
// Kernel 1: Q = x_mic @ w_mic^T + b_mic ; K = x_ref @ w_ref^T + b_ref
// Layout: [b][h][tp][fp], fp padded 129->132, t padded 500->512, pads = 0.
// Zero padding makes every WMMA fragment load in kernel 2 unconditional.
// ---------------------------------------------------------------------------
__global__ void qk_project_kernel(const float* __restrict__ x_mic,
                                  const float* __restrict__ x_ref,
                                  const float* __restrict__ w_mic,
                                  const float* __restrict__ b_mic,
                                  const float* __restrict__ w_ref,
                                  const float* __restrict__ b_ref,
                                  float* __restrict__ Q,
                                  float* __restrict__ K) {
    int idx = blockIdx.x * blockDim.x + threadIdx.x;
    const int N = B_ * H_ * TP_ * FP_;
    if (idx >= N) return;
    int fp = idx % FP_;
    int t  = (idx / FP_) % TP_;
    int h  = (idx / (FP_ * TP_)) % H_;
    int b  =  idx / (FP_ * TP_ * H_);
    float q = 0.f, k = 0.f;
    if (fp < F_ && t < T_) {
        int i0 = ((b * C_ + 0) * T_ + t) * F_ + fp;
        int i1 = ((b * C_ + 1) * T_ + t) * F_ + fp;
        float xm0 = x_mic[i0], xm1 = x_mic[i1];
        float xr0 = x_ref[i0], xr1 = x_ref[i1];
        q = w_mic[h * C_ + 0] * xm0 + w_mic[h * C_ + 1] * xm1 + b_mic[h];
        k = w_ref[h * C_ + 0] * xr0 + w_ref[h * C_ + 1] * xr1 + b_ref[h];
    }
    Q[idx] = q;
    K[idx] = k;
}

// ---------------------------------------------------------------------------
// Kernel 2: banded V[t,d] = sum_f Q[t,f] * K[t-99+d, f]  via fp32 WMMA.
// One wave per (bh, row-tile i). The 16x132 Q tile is held in 33 v2f register
// fragments and reused across the 8 column tiles j = i-7 .. i of the band.
// __launch_bounds__(256,1) raises the VGPR cap so the A-fragment cache does
// NOT spill to scratch (~120 live VGPRs/wave; wave32 budget is 1024).
// Two independent accumulator chains (even/odd K-chunks) double matrix-pipe
// ILP; combined with 8 scalar adds at the end.
// A-frag (16x4 f32): lane<16 -> M=lane&15, K=0/1 ; lane>=16 -> K=2/3.
// B-frag (4x16 f32, B = K^T): same pattern with N=lane&15 columns = t' rows.
// T padded to 512 with zeros -> all loads unconditional, EXEC all-1s at WMMA.
// ---------------------------------------------------------------------------
__global__ void __launch_bounds__(256, 1)
corr_wmma_kernel(const float* __restrict__ Q,
                 const float* __restrict__ K,
                 float* __restrict__ V) {
    const int wave = threadIdx.x >> 5;
    const int lane = threadIdx.x & 31;
    const int gw   = blockIdx.x * (blockDim.x >> 5) + wave;  // 0..1023
    const int bh   = gw >> 5;                 // 32 row tiles per (b,h)
    const int i    = gw & 31;                 // row tile: t in [16i, 16i+15]

    const float* __restrict__ Qb = Q + (size_t)bh * TP_ * FP_;
    const float* __restrict__ Kb = K + (size_t)bh * TP_ * FP_;

    const int l15 = lane & 15;
    const int hi  = lane >> 4;                // 0: K=0,1 ; 1: K=2,3
    const int tbase = i * 16;

    // Load the A (Q) tile fragments once; reuse across all band tiles.
    const float* qrow = Qb + (size_t)(tbase + l15) * FP_ + 2 * hi;
    v2f a[NF_];
    #pragma unroll
    for (int fc = 0; fc < NF_; ++fc)
        a[fc] = *(const v2f*)(qrow + 4 * fc);

    for (int jo = 0; jo < 8; ++jo) {
        int j = i - 7 + jo;                   // column tile (t' base / 16)
        if (j < 0) continue;                  // wave-uniform skip
        const int cbase = j * 16;
        const float* krow = Kb + (size_t)(cbase + l15) * FP_ + 2 * hi;

        v8f acc0 = {}, acc1 = {};
        #pragma unroll
        for (int fc = 0; fc + 1 < NF_; fc += 2) {
            v2f b0 = *(const v2f*)(krow + 4 * fc);
            v2f b1 = *(const v2f*)(krow + 4 * (fc + 1));
            acc0 = __builtin_amdgcn_wmma_f32_16x16x4_f32(
                false, a[fc], false, b0, (short)0, acc0, false, false);
            acc1 = __builtin_amdgcn_wmma_f32_16x16x4_f32(
                false, a[fc + 1], false, b1, (short)0, acc1, false, false);
        }
        {   // tail chunk (NF_ = 33 is odd)
            v2f b0 = *(const v2f*)(krow + 4 * (NF_ - 1));
            acc0 = __builtin_amdgcn_wmma_f32_16x16x4_f32(
                false, a[NF_ - 1], false, b0, (short)0, acc0, false, false);
        }

        // C/D layout: N = lane&15 (= t' offset), VGPR r -> M = r + 8*(lane>>4)
        const int tp = cbase + l15;           // t'
        #pragma unroll
        for (int r = 0; r < 8; ++r) {
            int t = tbase + r + 8 * hi;
            int d = tp - t + (D_ - 1);
            if (t < T_ && d >= 0 && d < D_)
                V[((size_t)bh * T_ + t) * D_ + d] = acc0[r] + acc1[r];
        }
    }
}

// ---------------------------------------------------------------------------
// Kernel 3: conv (sum over H, 5x3 window, pad t:(4,0), d:(1,1)) + bias,
// then softmax over d. One block per (b,t), thread = d.
// V entries with t' = tt-99+dd < 0 were never written -> treat as 0 (matches
// reference zero-padding of Ku).
// ---------------------------------------------------------------------------
__global__ void conv_softmax_kernel(const float* __restrict__ V,
                                    const float* __restrict__ w_conv,
                                    const float* __restrict__ b_conv,
                                    float* __restrict__ A) {
    const int b = blockIdx.x / T_;
    const int t = blockIdx.x % T_;
    const int d = threadIdx.x;                // blockDim.x == 128

    __shared__ float wc[H_ * 5 * 3];
    __shared__ float red[128];
    for (int k = threadIdx.x; k < H_ * 15; k += blockDim.x) wc[k] = w_conv[k];
    __syncthreads();

    float acc = -1e30f;
    if (d < D_) {
        acc = b_conv[0];
        for (int h = 0; h < H_; ++h) {
            const float* Vb = V + (size_t)(b * H_ + h) * T_ * D_;
            #pragma unroll
            for (int kt = 0; kt < 5; ++kt) {
                int tt = t - 4 + kt;
                if (tt < 0) continue;          // time zero-pad
                #pragma unroll
                for (int kd = 0; kd < 3; ++kd) {
                    int dd  = d - 1 + kd;
                    int tpr = tt - (D_ - 1) + dd;   // t' of that V entry
                    float val = (dd >= 0 && dd < D_ && tpr >= 0)
                                    ? Vb[(size_t)tt * D_ + dd] : 0.f;
                    acc += val * wc[(h * 5 + kt) * 3 + kd];
                }
            }
        }
    }

    // block softmax over d (inactive lanes carry -1e30 / 0)
    red[threadIdx.x] = acc;
    __syncthreads();
    for (int s = 64; s > 0; s >>= 1) {
        if (threadIdx.x < s)
            red[threadIdx.x] = fmaxf(red[threadIdx.x], red[threadIdx.x + s]);
        __syncthreads();
    }
    float m = red[0];
    __syncthreads();
    float e = (d < D_) ? __expf(acc - m) : 0.f;
    red[threadIdx.x] = e;
    __syncthreads();
    for (int s = 64; s > 0; s >>= 1) {
        if (threadIdx.x < s) red[threadIdx.x] += red[threadIdx.x + s];
        __syncthreads();
    }
    float den = red[0];
    if (d < D_)
        A[((size_t)b * T_ + t) * D_ + d] = e / den;
}

// ---------------------------------------------------------------------------
// Kernel 4: y[b,c,t,f] = sum_d x_ref[b,c,t-99+d,f] * A[b,t,d]
// One block per (b,t); A row staged in LDS; f-contiguous coalesced loads.
// ---------------------------------------------------------------------------
__global__ void apply_kernel(const float* __restrict__ x_ref,
                             const float* __restrict__ A,
                             float* __restrict__ y) {
    const int b = blockIdx.x / T_;
    const int t = blockIdx.x % T_;
    __shared__ float As[D_];
    for (int k = threadIdx.x; k < D_; k += blockDim.x)
        As[k] = A[((size_t)b * T_ + t) * D_ + k];
    __syncthreads();

    const int d0 = (t >= D_ - 1) ? 0 : (D_ - 1 - t);   // first d with t' >= 0
    for (int e = threadIdx.x; e < C_ * F_; e += blockDim.x) {
        int c = e / F_;
        int f = e % F_;
        const float* xr = x_ref + (size_t)(b * C_ + c) * T_ * F_ + f;
        float acc = 0.f;
        for (int d = d0; d < D_; ++d) {
            int tpr = t - (D_ - 1) + d;
            acc += xr[(size_t)tpr * F_] * As[d];
        }
        y[((size_t)(b * C_ + c) * T_ + t) * F_ + f] = acc;
    }
}

// ---------------------------------------------------------------------------
extern "C" void kernel_launch(void* const* d_in, const int* in_sizes, int n_in,
                              void* d_out, int out_size, void* d_ws, size_t ws_size,
                              hipStream_t stream) {
    (void)in_sizes; (void)n_in; (void)out_size; (void)ws_size;
    const float* x_mic  = (const float*)d_in[0];
    const float* x_ref  = (const float*)d_in[1];
    const float* w_mic  = (const float*)d_in[2];
    const float* b_mic  = (const float*)d_in[3];
    const float* w_ref  = (const float*)d_in[4];
    const float* b_ref  = (const float*)d_in[5];
    const float* w_conv = (const float*)d_in[6];
    const float* b_conv = (const float*)d_in[7];
    // d_in[8] = delay (==100), baked into D_.

    float* ws = (float*)d_ws;
    const size_t QK_ELEMS = (size_t)B_ * H_ * TP_ * FP_;   // 2,162,688
    const size_t V_ELEMS  = (size_t)B_ * H_ * T_ * D_;     // 1,600,000
    float* Q = ws;
    float* K = ws + QK_ELEMS;
    float* V = ws + 2 * QK_ELEMS;
    float* A = ws + 2 * QK_ELEMS + V_ELEMS;
    float* y = (float*)d_out;

    // 1) projections (F padded to 132, T padded to 512 with zeros)
    {
        int n = (int)QK_ELEMS;
        qk_project_kernel<<<(n + 255) / 256, 256, 0, stream>>>(
            x_mic, x_ref, w_mic, b_mic, w_ref, b_ref, Q, K);
    }
    // 2) banded QK^T via fp32 WMMA: 32 (b,h) x 32 row tiles = 1024 waves
    {
        int total_waves = B_ * H_ * 32;      // 1024
        corr_wmma_kernel<<<total_waves / 8, 256, 0, stream>>>(Q, K, V);
    }
    // 3) conv + softmax -> A
    conv_softmax_kernel<<<B_ * T_, 128, 0, stream>>>(V, w_conv, b_conv, A);
    // 4) weighted gather -> y
    apply_kernel<<<B_ * T_, 256, 0, stream>>>(x_ref, A, y);
}